// DBLM_79766132621682
// MI455X (gfx1250) — compile-verified
//
#include <hip/hip_runtime.h>

// Problem constants (from reference)
#define Bc   4
#define Tc   4
#define Nc   160
#define FINc 8
#define HTc  64
#define RTc  4
#define HSc  64
#define RSc  20
#define HHc  192
#define BTc  16   // B*T

typedef float v2f __attribute__((ext_vector_type(2)));
typedef float v8f __attribute__((ext_vector_type(8)));

enum { ACT_NONE = 0, ACT_RELU = 1, ACT_TANH = 2 };

// =====================================================================
// Generic batched fp32 WMMA GEMM:  D = act(alpha * A@B(+T) + beta*Cin + bias)
// One wave computes one 16x16 tile via V_WMMA_F32_16X16X4_F32.
// A: [M,K] row-major (lda), per-batch offset = (batch/divA)*strideA
// TRANSB=0 -> B [K,N] row-major (ldb); TRANSB=1 -> B [N,K] row-major (ldb)
// Cin/D: [M,N] row-major (ldd), per-batch stride strideD
// OOB rows/cols: addresses are CLAMPED (not predicated) so loads stay
// unconditional and EXEC remains all-1s; garbage affects only D elements
// that the guarded store never writes.
// =====================================================================
template <int TRANSB>
__global__ __launch_bounds__(128) void wmma_gemm_f32(
    const float* __restrict__ A, int lda, long long strideA, int divA,
    const float* __restrict__ Bm, int ldb, long long strideB,
    const float* __restrict__ Cin, float alpha, float beta,
    const float* __restrict__ bias,
    float* __restrict__ D, int ldd, long long strideD,
    int M, int Ncols, int K, int tilesN, int act)
{
  int wavesPerBlock = blockDim.x >> 5;
  int tile = blockIdx.x * wavesPerBlock + (threadIdx.x >> 5);
  int tilesM = (M + 15) >> 4;
  if (tile >= tilesM * tilesN) return;
  int batch = blockIdx.y;
  int mi = tile / tilesN;
  int ni = tile - mi * tilesN;

  const float* Ab = A + (long long)(batch / divA) * strideA;
  const float* Bb = Bm + (long long)batch * strideB;
  const float* Cb = Cin ? Cin + (long long)batch * strideD : (const float*)0;
  float* Db = D + (long long)batch * strideD;

  int lane = threadIdx.x & 31;
  int half = lane >> 4;     // 0: lanes 0-15, 1: lanes 16-31
  int idx  = lane & 15;
  int mrow = mi * 16 + idx; // A-matrix row for this lane
  int ncol = ni * 16 + idx; // B/C/D column for this lane
  bool nok = ncol < Ncols;

  // Clamp OOB lanes to a valid address (results discarded at store time).
  int mrowc = mrow < M ? mrow : M - 1;
  int ncolc = nok ? ncol : Ncols - 1;

  const float* Arow = Ab + (long long)mrowc * lda;                 // contiguous in k
  const float* Bcol = TRANSB ? Bb + (long long)ncolc * ldb         // contiguous in k
                             : Bb + ncolc;                         // stride ldb in k
  const int ka0 = half << 1;  // lanes 0-15 -> +0, lanes 16-31 -> +2

  v8f acc = {0.f, 0.f, 0.f, 0.f, 0.f, 0.f, 0.f, 0.f};

  int k = 0;
  // Main loop: 4 k-steps (K-chunk of 16) per iteration; loads clause, then 4 WMMAs.
  for (; k + 16 <= K; k += 16) {
    v2f av[4], bv[4];
#pragma unroll
    for (int u = 0; u < 4; ++u) {
      int ka = k + 4 * u + ka0;
      av[u] = *(const v2f*)(Arow + ka);
      if (TRANSB) {
        bv[u] = *(const v2f*)(Bcol + ka);
      } else {
        bv[u].x = Bcol[(long long)ka * ldb];
        bv[u].y = Bcol[(long long)(ka + 1) * ldb];
      }
    }
#pragma unroll
    for (int u = 0; u < 4; ++u) {
      acc = __builtin_amdgcn_wmma_f32_16x16x4_f32(
          false, av[u], false, bv[u], (short)0, acc, false, false);
    }
  }
  // Tail (K % 16, still multiple of 4)
  for (; k < K; k += 4) {
    int ka = k + ka0;
    v2f av = *(const v2f*)(Arow + ka);
    v2f bv;
    if (TRANSB) {
      bv = *(const v2f*)(Bcol + ka);
    } else {
      bv.x = Bcol[(long long)ka * ldb];
      bv.y = Bcol[(long long)(ka + 1) * ldb];
    }
    acc = __builtin_amdgcn_wmma_f32_16x16x4_f32(
        false, av, false, bv, (short)0, acc, false, false);
  }

  if (!nok) return;
  float badd = bias ? bias[ncol] : 0.f;
#pragma unroll
  for (int v = 0; v < 8; ++v) {
    int m = mi * 16 + v + (half << 3);  // D row: VGPR v -> M=v (lanes 0-15), v+8 (16-31)
    if (m < M) {
      float val = alpha * acc[v] + badd;
      if (Cb) val += beta * Cb[(long long)m * ldd + ncol];
      if (act == ACT_RELU)      val = fmaxf(val, 0.f);
      else if (act == ACT_TANH) val = tanhf(val);
      Db[(long long)m * ldd + ncol] = val;
    }
  }
}

// =====================================================================
// btcn: gated temporal conv, fwd + node-reversed bwd branch.
// H[t=0]=relu(fwd@0), H[1]=relu(fwd@1), H[2]=relu(bwd@1), H[3]=relu(bwd@0)
// weights: w[co][ci][0][k], flat co*cin*3 + ci*3 + k
// =====================================================================
__global__ void btcn_kernel(const float* __restrict__ Xin, float* __restrict__ Hout,
    const float* __restrict__ w1,  const float* __restrict__ b1,
    const float* __restrict__ w2,  const float* __restrict__ b2,
    const float* __restrict__ w3,  const float* __restrict__ b3,
    const float* __restrict__ w1b, const float* __restrict__ b1b,
    const float* __restrict__ w2b, const float* __restrict__ b2b,
    const float* __restrict__ w3b, const float* __restrict__ b3b,
    int cin, int cout)
{
  int gid = blockIdx.x * blockDim.x + threadIdx.x;
  int total = Bc * Nc * cout;
  if (gid >= total) return;
  int co = gid % cout;
  int n  = (gid / cout) % Nc;
  int bb = gid / (cout * Nc);

  for (int t0 = 0; t0 < 2; ++t0) {
    float a1 = b1[co],  a2 = b2[co],  a3 = b3[co];
    float c1 = b1b[co], c2 = b2b[co], c3 = b3b[co];
    for (int k = 0; k < 3; ++k) {
      const float* xf = Xin + (((long long)bb * Tc + (t0 + k)) * Nc + n) * cin;
      const float* xb = Xin + (((long long)bb * Tc + (t0 + k)) * Nc + (Nc - 1 - n)) * cin;
      long long wb = (long long)co * cin * 3 + k;
      for (int ci = 0; ci < cin; ++ci) {
        float xfv = xf[ci], xbv = xb[ci];
        long long wi = wb + (long long)ci * 3;
        a1 += xfv * w1[wi];  a2 += xfv * w2[wi];  a3 += xfv * w3[wi];
        c1 += xbv * w1b[wi]; c2 += xbv * w2b[wi]; c3 += xbv * w3b[wi];
      }
    }
    float of = a1 * (1.0f / (1.0f + expf(-a2))) + a3;
    float ob = c1 * (1.0f / (1.0f + expf(-c2))) + c3;
    Hout[(((long long)bb * Tc + t0) * Nc + n) * cout + co]       = fmaxf(of, 0.f);
    Hout[(((long long)bb * Tc + (3 - t0)) * Nc + n) * cout + co] = fmaxf(ob, 0.f);
  }
}

// Interleave Chebyshev terms: xcat[r, f*4 + c] = x_c[r, f]
__global__ void cheb_pack_kernel(const float* __restrict__ x0, const float* __restrict__ x1,
                                 const float* __restrict__ x2, const float* __restrict__ x3,
                                 float* __restrict__ xcat, int F, int total)
{
  int i = blockIdx.x * blockDim.x + threadIdx.x;
  if (i >= total) return;
  int f = i % F;
  long long r = i / F;
  long long o = r * (4LL * F) + f * 4;
  xcat[o + 0] = x0[i]; xcat[o + 1] = x1[i]; xcat[o + 2] = x2[i]; xcat[o + 3] = x3[i];
}

// ST = concat([H, E], -1) : rows = B*T*N, 64 + 64 -> 128
__global__ void concat_st_kernel(const float* __restrict__ H, const float* __restrict__ E,
                                 float* __restrict__ ST, int total)
{
  int i = blockIdx.x * blockDim.x + threadIdx.x;
  if (i >= total) return;
  int c = i & 127;
  long long r = i >> 7;
  ST[i] = (c < 64) ? H[r * 64 + c] : E[r * 64 + (c - 64)];
}

// v[k] = sum_c (a_w[c]+a_w[c+192]) * Wb_w[c][k]  (k<128) ; v[128] = aw . Wb_b
__global__ void fuse_prep_kernel(const float* __restrict__ Wb_w, const float* __restrict__ Wb_b,
                                 const float* __restrict__ a_w, float* __restrict__ v)
{
  int k = threadIdx.x;
  if (k < 128) {
    float s = 0.f;
    for (int c = 0; c < HHc; ++c) s += (a_w[c] + a_w[c + HHc]) * Wb_w[(long long)c * 128 + k];
    v[k] = s;
  }
  if (k == 0) {
    float s = 0.f;
    for (int c = 0; c < HHc; ++c) s += (a_w[c] + a_w[c + HHc]) * Wb_b[c];
    v[128] = s;
  }
}

// g[row] = const + ST[row,:] . v   (rows = BT*N = 2560)
__global__ void g_kernel(const float* __restrict__ ST, const float* __restrict__ v,
                         float* __restrict__ g, int rows)
{
  int r = blockIdx.x * blockDim.x + threadIdx.x;
  if (r >= rows) return;
  float s = v[128];
  const float* st = ST + (long long)r * 128;
  for (int k = 0; k < 128; ++k) s += st[k] * v[k];
  g[r] = s;
}

// attn row: e[i,j] = lrelu(g[(2i + (j>=80)) mod 160]); mask by adj>0 with NEG; softmax over j
__global__ void attn_kernel(const float* __restrict__ g, const float* __restrict__ adj,
                            float* __restrict__ attn)
{
  int row = blockIdx.x * blockDim.x + threadIdx.x;  // bt*160 + i
  if (row >= BTc * Nc) return;
  int i  = row % Nc;
  int bt = row / Nc;
  int b  = bt / Tc;
  int r0 = 2 * i;     if (r0 >= Nc) r0 -= Nc;
  int r1 = 2 * i + 1; if (r1 >= Nc) r1 -= Nc;
  float g0 = g[bt * Nc + r0], g1 = g[bt * Nc + r1];
  float e0 = g0 > 0.f ? g0 : 0.2f * g0;
  float e1 = g1 > 0.f ? g1 : 0.2f * g1;
  const float* arow = adj + ((long long)b * Nc + i) * Nc;
  float* orow = attn + ((long long)bt * Nc + i) * Nc;
  const float NEGV = -9.0e15f;
  float m = -3.4e38f;
  for (int j = 0; j < Nc; ++j) {
    float ev  = (j < 80) ? e0 : e1;
    float val = arow[j] > 0.f ? ev : NEGV;
    m = fmaxf(m, val);
  }
  float ssum = 0.f;
  for (int j = 0; j < Nc; ++j) {
    float ev  = (j < 80) ? e0 : e1;
    float val = arow[j] > 0.f ? ev : NEGV;
    float ex = expf(val - m);
    orow[j] = ex;
    ssum += ex;
  }
  float inv = 1.0f / ssum;
  for (int j = 0; j < Nc; ++j) orow[j] *= inv;
}

// =====================================================================
// Host side
// =====================================================================
static void launch_gemm(hipStream_t s,
    const float* A, int lda, long long sA, int divA,
    const float* Bm, int ldb, long long sB, int transB,
    const float* Cin, float alpha, float beta, const float* bias,
    float* D, int ldd, long long sD,
    int M, int Ncols, int K, int batch, int act)
{
  int tilesM = (M + 15) / 16, tilesN = (Ncols + 15) / 16;
  int tiles = tilesM * tilesN;
  const int waves = 4;
  dim3 grid((tiles + waves - 1) / waves, batch);
  if (transB)
    wmma_gemm_f32<1><<<grid, dim3(32 * waves), 0, s>>>(
        A, lda, sA, divA, Bm, ldb, sB, Cin, alpha, beta, bias,
        D, ldd, sD, M, Ncols, K, tilesN, act);
  else
    wmma_gemm_f32<0><<<grid, dim3(32 * waves), 0, s>>>(
        A, lda, sA, divA, Bm, ldb, sB, Cin, alpha, beta, bias,
        D, ldd, sD, M, Ncols, K, tilesN, act);
}

static void dgcn_layer(hipStream_t s, const float* adj, const float* Xin, int F,
                       const float* theta, const float* bias, int cout, int act,
                       float* x1, float* x2, float* x3, float* xcat, float* Eout)
{
  long long zs = (long long)Nc * F;  // per-(b,t) slice stride of z
  // x1 = A @ x0
  launch_gemm(s, adj, Nc, (long long)Nc * Nc, Tc, Xin, F, zs, 0,
              nullptr, 1.f, 0.f, nullptr, x1, F, zs, Nc, F, Nc, BTc, ACT_NONE);
  // x2 = 2*A@x1 - x0
  launch_gemm(s, adj, Nc, (long long)Nc * Nc, Tc, x1, F, zs, 0,
              Xin, 2.f, -1.f, nullptr, x2, F, zs, Nc, F, Nc, BTc, ACT_NONE);
  // x3 = 2*A@x2 - x1
  launch_gemm(s, adj, Nc, (long long)Nc * Nc, Tc, x2, F, zs, 0,
              x1, 2.f, -1.f, nullptr, x3, F, zs, Nc, F, Nc, BTc, ACT_NONE);
  int total = BTc * Nc * F;
  cheb_pack_kernel<<<(total + 255) / 256, 256, 0, s>>>(Xin, x1, x2, x3, xcat, F, total);
  // E = act(xcat @ theta + bias)  : M = BT*N, K = 4F, N = cout
  launch_gemm(s, xcat, 4 * F, 0, 1, theta, cout, 0, 0,
              nullptr, 1.f, 0.f, bias, Eout, cout, 0, BTc * Nc, cout, 4 * F, 1, act);
}

struct TcnP {
  const float *w1, *b1, *w2, *b2, *w3, *b3, *w1b, *b1b, *w2b, *b2b, *w3b, *b3b;
};

extern "C" void kernel_launch(void* const* d_in, const int* in_sizes, int n_in,
                              void* d_out, int out_size, void* d_ws, size_t ws_size,
                              hipStream_t stream)
{
  auto F = [&](int i) { return (const float*)d_in[i]; };
  const float* X   = F(0);  // (B,T,N,8)
  const float* adj = F(1);  // (B,N,N)

  TcnP tc[3];
  const float *th[3], *bi[3];
  const float *Wb_w, *Wb_b, *Wfu_w, *a_w;

  // Detect flattening order: jax tree_flatten sorts dict keys -> fus.Wb_b (192) at index 2;
  // plain insertion order -> tcn1.w1 (1536) at index 2.
  bool sorted = (n_in >= 3 && in_sizes[2] == HHc);
  if (sorted) {
    Wb_b = F(2); Wb_w = F(3); Wfu_w = F(4); a_w = F(5);
    bi[0] = F(6);  th[0] = F(7);
    bi[1] = F(8);  th[1] = F(9);
    bi[2] = F(10); th[2] = F(11);
    for (int t = 0; t < 3; ++t) {
      int s = 12 + 12 * t;  // keys: b1,b1b,b2,b2b,b3,b3b,w1,w1b,w2,w2b,w3,w3b
      tc[t].b1  = F(s + 0); tc[t].b1b = F(s + 1);
      tc[t].b2  = F(s + 2); tc[t].b2b = F(s + 3);
      tc[t].b3  = F(s + 4); tc[t].b3b = F(s + 5);
      tc[t].w1  = F(s + 6); tc[t].w1b = F(s + 7);
      tc[t].w2  = F(s + 8); tc[t].w2b = F(s + 9);
      tc[t].w3  = F(s + 10); tc[t].w3b = F(s + 11);
    }
  } else {
    for (int t = 0; t < 3; ++t) {
      int s = 2 + 12 * t;   // keys: w1,b1,w2,b2,w3,b3,w1b,b1b,w2b,b2b,w3b,b3b
      tc[t].w1  = F(s + 0); tc[t].b1  = F(s + 1);
      tc[t].w2  = F(s + 2); tc[t].b2  = F(s + 3);
      tc[t].w3  = F(s + 4); tc[t].b3  = F(s + 5);
      tc[t].w1b = F(s + 6); tc[t].b1b = F(s + 7);
      tc[t].w2b = F(s + 8); tc[t].b2b = F(s + 9);
      tc[t].w3b = F(s + 10); tc[t].b3b = F(s + 11);
    }
    th[0] = F(38); bi[0] = F(39);
    th[1] = F(40); bi[1] = F(41);
    th[2] = F(42); bi[2] = F(43);
    Wb_w = F(44); Wb_b = F(45); Wfu_w = F(46); a_w = F(47);
  }

  // Workspace layout (floats)
  float* w = (float*)d_ws;
  size_t off = 0;
  auto alloc = [&](size_t n) { float* p = w + off; off += n; return p; };
  float* H1   = alloc((size_t)BTc * Nc * HTc);   // 163840
  float* H2   = alloc((size_t)BTc * Nc * RTc);   // 10240
  float* H3   = alloc((size_t)BTc * Nc * HTc);   // 163840 (final H)
  float* x1   = alloc((size_t)BTc * Nc * 64);
  float* x2   = alloc((size_t)BTc * Nc * 64);
  float* x3   = alloc((size_t)BTc * Nc * 64);
  float* xcat = alloc((size_t)BTc * Nc * 256);
  float* E1   = alloc((size_t)BTc * Nc * HSc);
  float* E2   = alloc((size_t)BTc * Nc * RSc);
  float* E3   = alloc((size_t)BTc * Nc * HSc);
  float* STb  = alloc((size_t)BTc * Nc * 128);
  float* vv   = alloc(132);
  float* gg   = alloc((size_t)BTc * Nc);
  float* PT   = alloc((size_t)BTc * Nc * Nc);
  float* Q    = alloc((size_t)BTc * Nc * Nc);
  float* AT   = alloc((size_t)BTc * Nc * Nc);
  (void)ws_size; (void)out_size;

  // ---- btcn chain: 8 -> 64 -> 4 -> 64 ----
  {
    int t1 = Bc * Nc * HTc;
    btcn_kernel<<<(t1 + 127) / 128, 128, 0, stream>>>(X, H1,
        tc[0].w1, tc[0].b1, tc[0].w2, tc[0].b2, tc[0].w3, tc[0].b3,
        tc[0].w1b, tc[0].b1b, tc[0].w2b, tc[0].b2b, tc[0].w3b, tc[0].b3b, FINc, HTc);
    int t2 = Bc * Nc * RTc;
    btcn_kernel<<<(t2 + 127) / 128, 128, 0, stream>>>(H1, H2,
        tc[1].w1, tc[1].b1, tc[1].w2, tc[1].b2, tc[1].w3, tc[1].b3,
        tc[1].w1b, tc[1].b1b, tc[1].w2b, tc[1].b2b, tc[1].w3b, tc[1].b3b, HTc, RTc);
    int t3 = Bc * Nc * HTc;
    btcn_kernel<<<(t3 + 127) / 128, 128, 0, stream>>>(H2, H3,
        tc[2].w1, tc[2].b1, tc[2].w2, tc[2].b2, tc[2].w3, tc[2].b3,
        tc[2].w1b, tc[2].b1b, tc[2].w2b, tc[2].b2b, tc[2].w3b, tc[2].b3b, RTc, HTc);
  }

  // ---- dgcn chain ----
  dgcn_layer(stream, adj, X,  FINc, th[0], bi[0], HSc, ACT_RELU, x1, x2, x3, xcat, E1);
  dgcn_layer(stream, adj, E1, HSc,  th[1], bi[1], RSc, ACT_NONE, x1, x2, x3, xcat, E2);
  dgcn_layer(stream, adj, E2, RSc,  th[2], bi[2], HSc, ACT_RELU, x1, x2, x3, xcat, E3);

  // ---- fusion ----
  {
    int totST = BTc * Nc * 128;
    concat_st_kernel<<<(totST + 255) / 256, 256, 0, stream>>>(H3, E3, STb, totST);

    fuse_prep_kernel<<<1, 128, 0, stream>>>(Wb_w, Wb_b, a_w, vv);

    int rows = BTc * Nc;
    g_kernel<<<(rows + 127) / 128, 128, 0, stream>>>(STb, vv, gg, rows);
    attn_kernel<<<(rows + 127) / 128, 128, 0, stream>>>(gg, adj, AT);

    // P_tmp = ST @ Wfu_w^T : (2560 x 128) @ (128 x 160)
    launch_gemm(stream, STb, 128, 0, 1, Wfu_w, 128, 0, 1,
                nullptr, 1.f, 0.f, nullptr, PT, Nc, 0, BTc * Nc, Nc, 128, 1, ACT_NONE);
    // Q = adjr @ P_tmp : 16 batches of (160x160)@(160x160); adj shared across t
    launch_gemm(stream, adj, Nc, (long long)Nc * Nc, Tc, PT, Nc, (long long)Nc * Nc, 0,
                nullptr, 1.f, 0.f, nullptr, Q, Nc, (long long)Nc * Nc, Nc, Nc, Nc, BTc, ACT_NONE);
    // out = tanh(attn @ Q) -> d_out (B,T,N,N)
    launch_gemm(stream, AT, Nc, (long long)Nc * Nc, 1, Q, Nc, (long long)Nc * Nc, 0,
                nullptr, 1.f, 0.f, nullptr, (float*)d_out, Nc, (long long)Nc * Nc,
                Nc, Nc, Nc, BTc, ACT_TANH);
  }
}